// Decoder_39625368272934
// MI455X (gfx1250) — compile-verified
//
#include <hip/hip_runtime.h>
#include <math.h>

// ---------------------------------------------------------------------------
// CDNA5 (gfx1250) fp32 WMMA path: V_WMMA_F32_16X16X4_F32
//   A: 16x4 f32  -> 2 VGPRs/lane: lanes 0-15 hold M=lane (K=k0 in v0, k0+1 in
//      v1); lanes 16-31 hold M=lane-16 (K=k0+2 in v0, k0+3 in v1)
//   B: 4x16 f32  -> mirror layout over N
//   C/D: 16x16 f32 -> 8 VGPRs; vgpr j = row (j + 8*(lane>=16)), col lane%16
// ---------------------------------------------------------------------------
typedef __attribute__((ext_vector_type(2))) float v2f;
typedef __attribute__((ext_vector_type(8))) float v8f;

#define WMMA_F32X4(a, b, c) \
  __builtin_amdgcn_wmma_f32_16x16x4_f32(false, (a), false, (b), (short)0, (c), false, false)

#define NP 512
#define BK 32   // k-rows staged to LDS per chunk in the attention kernel

// ---------------------------------------------------------------------------
// Generic tile GEMM: C(M,N) = A(M,K) * B(N,K)^T, A/B/C row-major.
// One wave (32 threads) per 16x16 output tile. M,N multiples of 16; K
// remainder (e.g. K=1026, K%4==2) zero-padded.
// ---------------------------------------------------------------------------
__global__ __launch_bounds__(32) void gemm_abt(const float* __restrict__ A,
                                               const float* __restrict__ B,
                                               float* __restrict__ C,
                                               int M, int N, int K) {
  const int tileN = blockIdx.x, tileM = blockIdx.y;
  const int lane = threadIdx.x;
  const int half = lane >> 4;
  const int l16  = lane & 15;
  const int m = tileM * 16 + l16;
  const int n = tileN * 16 + l16;
  const float* arow = A + (size_t)m * K + (half ? 2 : 0);
  const float* brow = B + (size_t)n * K + (half ? 2 : 0);

  v8f acc = {};
  int k = 0;
  for (; k + 4 <= K; k += 4) {
    v2f av, bv;
    av.x = arow[k]; av.y = arow[k + 1];
    bv.x = brow[k]; bv.y = brow[k + 1];
    acc = WMMA_F32X4(av, bv, acc);
  }
  if (k < K) {                      // remainder (K%4 == 1..3), zero-pad
    v2f av = {}, bv = {};
    const int kk = k + (half ? 2 : 0);
    if (kk < K)     { av.x = A[(size_t)m * K + kk];     bv.x = B[(size_t)n * K + kk]; }
    if (kk + 1 < K) { av.y = A[(size_t)m * K + kk + 1]; bv.y = B[(size_t)n * K + kk + 1]; }
    acc = WMMA_F32X4(av, bv, acc);
  }
  const int col = tileN * 16 + l16;
#pragma unroll
  for (int j = 0; j < 8; ++j) {
    const int row = tileM * 16 + half * 8 + j;
    C[(size_t)row * N + col] = acc[j];
  }
}

// ---------------------------------------------------------------------------
// Fused attention energy:
//   e[b, l] = sum_p nu[p] * tanh( upred[b,p] + uf[p,l] + sum_c Ua[p,c]*a[b,c,l] )
// Block = 256 threads (8 waves) owns (b, 16 l-columns); wave w does p-tiles
// 4w..4w+3 of the 32 p-tiles (NP=512).
//
// The B slab a[b, :, l0:l0+16] is staged into LDS ONCE per block (vs 8x
// redundant per-wave global loads) with CDNA5 async-to-LDS copies, double
// buffered on ASYNCcnt, and stored TRANSPOSED [l][k] so each lane's B pair is
// one contiguous ds_load_b64. u_a never touches memory.
// ---------------------------------------------------------------------------
__global__ __launch_bounds__(256) void attn_e_kernel(
    const float* __restrict__ Ua,     // (NP, C)
    const float* __restrict__ a,      // (B, C, L)
    const float* __restrict__ upred,  // (B, NP)
    const float* __restrict__ uf,     // (NP, L)
    const float* __restrict__ nu,     // (NP)
    float* __restrict__ e,            // (B, L)
    int C, int L) {
  const int b    = blockIdx.y;
  const int l0   = blockIdx.x * 16;
  const int tid  = threadIdx.x;
  const int wave = tid >> 5;
  const int lane = tid & 31;
  const int half = lane >> 4;
  const int l16  = lane & 15;
  const float* ab = a + (size_t)b * C * L;

  __shared__ float sB[2][16 * BK];    // transposed: sB[buf][l*BK + k]
  __shared__ float epart[8][16];

  const unsigned sB0 = (unsigned)(uintptr_t)(&sB[0][0]);  // LDS byte offset
  const int nch = (C + BK - 1) / BK;

  // Stage chunk -> LDS buffer via async DMA. 512 elements / 256 threads = 2
  // asyncs per thread, always issued (addresses clamped) so ASYNCcnt math is
  // exact and EXEC stays uniform.
  auto stage = [&](int chunk, int buf) {
#pragma unroll
    for (int rep = 0; rep < 2; ++rep) {
      const int el = tid + rep * 256;        // 0..511
      const int kl = el >> 4;                // 0..31 (k within chunk)
      const int l  = el & 15;                // 0..15
      int gr = chunk * BK + kl;
      if (gr > C - 1) gr = C - 1;            // clamp (tail zero-guarded later)
      const unsigned lofs = sB0 + (unsigned)(buf * (16 * BK) + l * BK + kl) * 4u;
      const unsigned long long ga =
          (unsigned long long)(uintptr_t)(ab + (size_t)gr * L + l0 + l);
      asm volatile("global_load_async_to_lds_b32 %0, %1, off"
                   :: "v"(lofs), "v"(ga) : "memory");
    }
  };

  v8f acc[4] = {};
  stage(0, 0);
  for (int c = 0; c < nch; ++c) {
    if (c + 1 < nch) {
      stage(c + 1, (c + 1) & 1);             // prefetch next chunk
      asm volatile("s_wait_asynccnt 0x2" ::: "memory");  // chunk c retired
    } else {
      asm volatile("s_wait_asynccnt 0x0" ::: "memory");
    }
    __syncthreads();                         // chunk c visible to all waves

    const float* bufp = &sB[c & 1][0];
    const int k0   = c * BK;
    const int kmax = C - k0;                 // valid rows in this chunk
    if (kmax >= BK) {                        // full chunk: 8 k-steps
#pragma unroll
      for (int ks = 0; ks < BK / 4; ++ks) {
        const int kb = ks * 4 + half * 2;    // local k for this half
        const v2f bv = *(const v2f*)&bufp[l16 * BK + kb];   // ds_load_b64
#pragma unroll
        for (int t = 0; t < 4; ++t) {
          const int p = (wave * 4 + t) * 16 + l16;
          const float* ar = Ua + (size_t)p * C + k0 + kb;
          v2f av;
          av.x = ar[0]; av.y = ar[1];
          acc[t] = WMMA_F32X4(av, bv, acc[t]);
        }
      }
    } else {                                 // tail chunk: zero-pad K
      const int ksteps = (kmax + 3) >> 2;
      for (int ks = 0; ks < ksteps; ++ks) {
        const int kb = ks * 4 + half * 2;
        v2f bv = {};
        if (kb < kmax)     bv.x = bufp[l16 * BK + kb];
        if (kb + 1 < kmax) bv.y = bufp[l16 * BK + kb + 1];
#pragma unroll
        for (int t = 0; t < 4; ++t) {
          const int p = (wave * 4 + t) * 16 + l16;
          v2f av = {};
          if (kb < kmax)     av.x = Ua[(size_t)p * C + k0 + kb];
          if (kb + 1 < kmax) av.y = Ua[(size_t)p * C + k0 + kb + 1];
          acc[t] = WMMA_F32X4(av, bv, acc[t]);
        }
      }
    }
    __syncthreads();                         // done reading buf before restage
  }

  // epilogue: add upred/uf, tanh, weight by nu, reduce over p
  float esum = 0.f;
#pragma unroll
  for (int t = 0; t < 4; ++t) {
#pragma unroll
    for (int j = 0; j < 8; ++j) {
      const int p = (wave * 4 + t) * 16 + half * 8 + j;
      const float v = acc[t][j] + upred[(size_t)b * NP + p] + uf[(size_t)p * L + l0 + l16];
      esum += nu[p] * tanhf(v);
    }
  }
  esum += __shfl_xor(esum, 16, 32);          // fold the two half-lane groups
  if (lane < 16) epart[wave][lane] = esum;
  __syncthreads();
  if (tid < 16) {
    float s = 0.f;
#pragma unroll
    for (int w = 0; w < 8; ++w) s += epart[w][tid];
    e[(size_t)b * L + l0 + tid] = s;
  }
}

// ---------------------------------------------------------------------------
// out[r] = dot(W[r,:], v) + bias[r]   (one wave per row; float4 loads so the
// 340MB/67MB fc_w streams can saturate HBM)
// ---------------------------------------------------------------------------
__global__ __launch_bounds__(256) void matvec_rows(const float* __restrict__ W,
                                                   const float* __restrict__ bias,
                                                   const float* __restrict__ v,
                                                   float* __restrict__ out,
                                                   int rows, int len) {
  const int row = blockIdx.x * 8 + (threadIdx.x >> 5);
  const int lane = threadIdx.x & 31;
  if (row >= rows) return;
  const float4* wr = (const float4*)(W + (size_t)row * len);
  const float4* vv = (const float4*)v;
  const int n4 = len >> 2;                   // len is a multiple of 4
  float s = 0.f;
  for (int i = lane; i < n4; i += 32) {
    const float4 a = wr[i], b = vv[i];
    s += a.x * b.x + a.y * b.y + a.z * b.z + a.w * b.w;
  }
  for (int m = 16; m >= 1; m >>= 1) s += __shfl_xor(s, m, 32);
  if (lane == 0) out[row] = s + bias[row];
}

// ---------------------------------------------------------------------------
// softmax over last dim, one block per batch row
// ---------------------------------------------------------------------------
__global__ __launch_bounds__(256) void softmax_kernel(const float* __restrict__ e,
                                                      float* __restrict__ alpha, int L) {
  const int b = blockIdx.x;
  const float* er = e + (size_t)b * L;
  float* ar = alpha + (size_t)b * L;
  __shared__ float red[256];
  float mx = -1e30f;
  for (int i = threadIdx.x; i < L; i += 256) mx = fmaxf(mx, er[i]);
  red[threadIdx.x] = mx; __syncthreads();
  for (int s = 128; s > 0; s >>= 1) {
    if (threadIdx.x < s) red[threadIdx.x] = fmaxf(red[threadIdx.x], red[threadIdx.x + s]);
    __syncthreads();
  }
  mx = red[0]; __syncthreads();
  float sum = 0.f;
  for (int i = threadIdx.x; i < L; i += 256) { float v = __expf(er[i] - mx); ar[i] = v; sum += v; }
  red[threadIdx.x] = sum; __syncthreads();
  for (int s = 128; s > 0; s >>= 1) {
    if (threadIdx.x < s) red[threadIdx.x] += red[threadIdx.x + s];
    __syncthreads();
  }
  const float inv = 1.f / red[0];
  for (int i = threadIdx.x; i < L; i += 256) ar[i] *= inv;
}

// ---------------------------------------------------------------------------
// ctx[b, off+c] = sum_l alpha[b,l] * a[b,c,l]   (one wave per channel, float4)
// ---------------------------------------------------------------------------
__global__ __launch_bounds__(256) void ctx_kernel(const float* __restrict__ alpha,
                                                  const float* __restrict__ a,
                                                  float* __restrict__ ctx,
                                                  int C, int L, int stride, int off) {
  const int b = blockIdx.y;
  const int c = blockIdx.x * 8 + (threadIdx.x >> 5);
  const int lane = threadIdx.x & 31;
  if (c >= C) return;
  const float4* ar = (const float4*)(a + ((size_t)b * C + c) * L);
  const float4* al = (const float4*)(alpha + (size_t)b * L);
  const int n4 = L >> 2;
  float s = 0.f;
  for (int i = lane; i < n4; i += 32) {
    const float4 x = ar[i], y = al[i];
    s += x.x * y.x + x.y * y.y + x.z * y.z + x.w * y.w;
  }
  for (int m = 16; m >= 1; m >>= 1) s += __shfl_xor(s, m, 32);
  if (lane == 0) ctx[(size_t)b * stride + off + c] = s;
}

// ---------------------------------------------------------------------------
// GRU gate fuse: gi/gh are (B,768) = [r|z|n] gates; out = (1-z)*n + z*h
// ---------------------------------------------------------------------------
__global__ __launch_bounds__(256) void gru_fuse(const float* __restrict__ gi,
                                                const float* __restrict__ gh,
                                                const float* __restrict__ b_ih,
                                                const float* __restrict__ b_hh,
                                                const float* __restrict__ h,
                                                float* __restrict__ out) {
  const int idx = blockIdx.x * 256 + threadIdx.x;  // 64*256
  const int b = idx >> 8, j = idx & 255;
  const float ir  = gi[b * 768 + j]       + b_ih[j];
  const float iz  = gi[b * 768 + 256 + j] + b_ih[256 + j];
  const float in_ = gi[b * 768 + 512 + j] + b_ih[512 + j];
  const float hr  = gh[b * 768 + j]       + b_hh[j];
  const float hz  = gh[b * 768 + 256 + j] + b_hh[256 + j];
  const float hn  = gh[b * 768 + 512 + j] + b_hh[512 + j];
  const float r = 1.f / (1.f + __expf(-(ir + hr)));
  const float z = 1.f / (1.f + __expf(-(iz + hz)));
  const float n = tanhf(in_ + r * hn);
  out[idx] = (1.f - z) * n + z * h[idx];
}

__global__ __launch_bounds__(256) void gather_embed(const int* __restrict__ x,
                                                    const float* __restrict__ emb,
                                                    float* __restrict__ out) {
  const int idx = blockIdx.x * 256 + threadIdx.x;  // 64*256
  const int b = idx >> 8, j = idx & 255;
  out[idx] = emb[(size_t)x[b] * 256 + j];
}

__global__ __launch_bounds__(256) void add3_kernel(const float* __restrict__ a,
                                                   const float* __restrict__ b,
                                                   const float* __restrict__ c,
                                                   float* __restrict__ out) {
  const int idx = blockIdx.x * 256 + threadIdx.x;
  out[idx] = a[idx] + b[idx] + c[idx];
}

// ---------------------------------------------------------------------------
extern "C" void kernel_launch(void* const* d_in, const int* in_sizes, int n_in,
                              void* d_out, int out_size, void* d_ws, size_t ws_size,
                              hipStream_t stream) {
  const int*   x          = (const int*)  d_in[0];
  const float* hidden     = (const float*)d_in[1];   // (1,64,256)
  const float* low_res    = (const float*)d_in[2];   // (64,684,256)
  const float* high_res   = (const float*)d_in[3];   // (64,342,576)
  const float* alpha_low  = (const float*)d_in[4];   // (1,256)
  const float* alpha_high = (const float*)d_in[5];   // (1,576)
  const float* embedding  = (const float*)d_in[6];
  const float* g1_wih     = (const float*)d_in[7];
  const float* g1_whh     = (const float*)d_in[8];
  const float* g1_bih     = (const float*)d_in[9];
  const float* g1_bhh     = (const float*)d_in[10];
  const float* g2_wih     = (const float*)d_in[11];
  const float* g2_whh     = (const float*)d_in[12];
  const float* g2_bih     = (const float*)d_in[13];
  const float* g2_bhh     = (const float*)d_in[14];
  const float* fcw_lo     = (const float*)d_in[15];
  const float* fcb_lo     = (const float*)d_in[16];
  const float* Up_lo      = (const float*)d_in[17];
  const float* Ua_lo      = (const float*)d_in[18];
  const float* Uf_lo      = (const float*)d_in[19];
  const float* nu_lo      = (const float*)d_in[20];
  const float* fcw_hi     = (const float*)d_in[21];
  const float* fcb_hi     = (const float*)d_in[22];
  const float* Up_hi      = (const float*)d_in[23];
  const float* Ua_hi      = (const float*)d_in[24];
  const float* Uf_hi      = (const float*)d_in[25];
  const float* nu_hi      = (const float*)d_in[26];
  const float* W_o        = (const float*)d_in[27];
  const float* W_s        = (const float*)d_in[28];
  const float* W_c        = (const float*)d_in[29];

  float* out   = (float*)d_out;              // (64,144)
  float* new_h = (float*)d_out + 64 * 144;   // (1,64,256)

  // workspace layout (floats); total ~1.16M floats (~4.7 MB)
  float* ws = (float*)d_ws;
  float* embedded = ws;                 ws += 64 * 256;
  float* gi1      = ws;                 ws += 64 * 768;
  float* gh1      = ws;                 ws += 64 * 768;
  float* pred     = ws;                 ws += 64 * 256;
  float* up_lo    = ws;                 ws += 64 * 512;
  float* up_hi    = ws;                 ws += 64 * 512;
  float* outf_lo  = ws;                 ws += 256 * 256;
  float* outf_hi  = ws;                 ws += 576 * 256;
  float* uf_lo    = ws;                 ws += 512 * 256;
  float* uf_hi    = ws;                 ws += 512 * 576;
  float* e_lo     = ws;                 ws += 64 * 256;
  float* e_hi     = ws;                 ws += 64 * 576;
  float* al_lo    = ws;                 ws += 64 * 256;
  float* al_hi    = ws;                 ws += 64 * 576;
  float* ctx      = ws;                 ws += 64 * 1026;
  float* gi2      = ws;                 ws += 64 * 768;
  float* gh2      = ws;                 ws += 64 * 768;
  float* tmp      = ws;                 ws += 64 * 256;
  float* wsv      = ws;                 ws += 64 * 256;
  float* wcv      = ws;                 ws += 64 * 256;

  // 1) embedding gather + GRU1 -> pred
  gather_embed<<<64, 256, 0, stream>>>(x, embedding, embedded);
  gemm_abt<<<dim3(48, 4), 32, 0, stream>>>(embedded, g1_wih, gi1, 64, 768, 256);
  gemm_abt<<<dim3(48, 4), 32, 0, stream>>>(hidden,   g1_whh, gh1, 64, 768, 256);
  gru_fuse<<<64, 256, 0, stream>>>(gi1, gh1, g1_bih, g1_bhh, hidden, pred);

  // 2) attention precomputes
  gemm_abt<<<dim3(32, 4), 32, 0, stream>>>(pred, Up_lo, up_lo, 64, 512, 256);
  gemm_abt<<<dim3(32, 4), 32, 0, stream>>>(pred, Up_hi, up_hi, 64, 512, 256);
  matvec_rows<<<65536 / 8, 256, 0, stream>>>(fcw_lo, fcb_lo, alpha_low,  outf_lo, 65536, 256);
  matvec_rows<<<147456 / 8, 256, 0, stream>>>(fcw_hi, fcb_hi, alpha_high, outf_hi, 147456, 576);
  gemm_abt<<<dim3(16, 32), 32, 0, stream>>>(Uf_lo, outf_lo, uf_lo, 512, 256, 256);
  gemm_abt<<<dim3(36, 32), 32, 0, stream>>>(Uf_hi, outf_hi, uf_hi, 512, 576, 256);

  // 3) fused energy (WMMA + async-LDS staging) -> softmax -> context
  attn_e_kernel<<<dim3(16, 64), 256, 0, stream>>>(Ua_lo, low_res,  up_lo, uf_lo, nu_lo, e_lo, 684, 256);
  attn_e_kernel<<<dim3(36, 64), 256, 0, stream>>>(Ua_hi, high_res, up_hi, uf_hi, nu_hi, e_hi, 342, 576);
  softmax_kernel<<<64, 256, 0, stream>>>(e_lo, al_lo, 256);
  softmax_kernel<<<64, 256, 0, stream>>>(e_hi, al_hi, 576);
  ctx_kernel<<<dim3(86, 64), 256, 0, stream>>>(al_lo, low_res,  ctx, 684, 256, 1026, 0);
  ctx_kernel<<<dim3(43, 64), 256, 0, stream>>>(al_hi, high_res, ctx, 342, 576, 1026, 684);

  // 4) GRU2 -> new_h (written straight into d_out)
  gemm_abt<<<dim3(48, 4), 32, 0, stream>>>(ctx,  g2_wih, gi2, 64, 768, 1026);
  gemm_abt<<<dim3(48, 4), 32, 0, stream>>>(pred, g2_whh, gh2, 64, 768, 256);
  gru_fuse<<<64, 256, 0, stream>>>(gi2, gh2, g2_bih, g2_bhh, pred, new_h);

  // 5) output head
  gemm_abt<<<dim3(16, 4), 32, 0, stream>>>(new_h, W_s, wsv, 64, 256, 256);
  gemm_abt<<<dim3(16, 4), 32, 0, stream>>>(ctx,   W_c, wcv, 64, 256, 1026);
  add3_kernel<<<64, 256, 0, stream>>>(embedded, wsv, wcv, tmp);
  gemm_abt<<<dim3(9, 4), 32, 0, stream>>>(tmp, W_o, out, 64, 144, 256);
}